// Model_12180527251634
// MI455X (gfx1250) — compile-verified
//
#include <hip/hip_runtime.h>
#include <hip/hip_bf16.h>

typedef __attribute__((ext_vector_type(2))) float v2f;
typedef __attribute__((ext_vector_type(8))) float v8f;

#define EPS 1e-5f

// ---- workspace layout (float offsets) --------------------------------------
// XN : normalized x, padded K 739 -> 768 with zeros   512 x 768
// H1 : relu(XN @ w1^T + bb1)                          512 x 128
// F  : post-attention, relu'd, reshaped               512 x 512
// FN : batchnorm(F)                                   512 x 512
// H2 : relu(FN @ w2^T + bb2)                          512 x 32
#define XN_LD   768
#define XN_OFF  0
#define H1_OFF  (512 * 768)
#define F_OFF   (H1_OFF + 512 * 128)
#define FN_OFF  (F_OFF + 512 * 512)
#define H2_OFF  (FN_OFF + 512 * 512)
// total floats = H2_OFF + 512*32 ~= 1.0M floats ~= 4 MB of d_ws

// ---- K1: batchnorm0 over batch axis, fused normalize, zero-pad K ----------
__global__ void bn0_normalize(const float* __restrict__ x,
                              const float* __restrict__ g,
                              const float* __restrict__ b,
                              float* __restrict__ XN) {
  int c = blockIdx.x * blockDim.x + threadIdx.x;   // 0..767
  if (c < 739) {
    float s = 0.f, s2 = 0.f;
    for (int r = 0; r < 512; ++r) {
      float v = x[r * 739 + c];
      s += v; s2 += v * v;
    }
    float m   = s  * (1.f / 512.f);
    float var = s2 * (1.f / 512.f) - m * m;
    float sc  = g[c] * rsqrtf(var + EPS);
    float bi  = b[c] - m * sc;
    for (int r = 0; r < 512; ++r)
      XN[r * XN_LD + c] = x[r * 739 + c] * sc + bi;
  } else if (c < 768) {
    for (int r = 0; r < 512; ++r) XN[r * XN_LD + c] = 0.f;  // zero K-pad
  }
}

// ---- K2: GEMM1 via V_WMMA_F32_16X16X4_F32, fused bias+relu ----------------
// H1(512x128) = relu( XN(512x739) @ w1^T(739x128) + bb1 )
// One wave per 16x16 output tile; 32x8 = 256 tiles.
// Main loop (K=0..735) is branch-free; single peeled tail step covers
// K=736..739 with the guard evaluated once (A pad columns in XN are zero).
__global__ void gemm1_wmma(const float* __restrict__ XN,
                           const float* __restrict__ w1,
                           const float* __restrict__ bb1,
                           float* __restrict__ H1) {
  int wave = (blockIdx.x * blockDim.x + threadIdx.x) >> 5;
  int lane = threadIdx.x & 31;
  int half = lane >> 4;          // 0: K={0,1}/M 0..7 ; 1: K={2,3}/M 8..15
  int lr   = lane & 15;
  int tm   = wave >> 3;          // 0..31
  int tn   = wave & 7;           // 0..7
  const float* arow = XN + (tm * 16 + lr) * XN_LD;  // A row, 8B aligned
  const float* brow = w1 + (tn * 16 + lr) * 739;    // B col of w1^T = row of w1
  v8f acc = {};
  for (int kk = 0; kk < 736; kk += 4) {             // 184 branch-free steps
    int k2 = kk + 2 * half;                         // k2+1 <= 735 < 739: safe
    v2f a = *reinterpret_cast<const v2f*>(arow + k2);
    v2f bv;
    bv.x = brow[k2];
    bv.y = brow[k2 + 1];
    acc = __builtin_amdgcn_wmma_f32_16x16x4_f32(false, a, false, bv,
                                                (short)0, acc, false, false);
  }
  {                                                 // tail: K = 736..739
    int k2 = 736 + 2 * half;                        // 736 or 738
    v2f a = *reinterpret_cast<const v2f*>(arow + k2);  // 739.. are zero pad
    v2f bv;
    bv.x = brow[k2];                                // 736 / 738: valid
    bv.y = half ? 0.f : brow[k2 + 1];               // 737 valid; 739 -> 0
    acc = __builtin_amdgcn_wmma_f32_16x16x4_f32(false, a, false, bv,
                                                (short)0, acc, false, false);
  }
  int col    = tn * 16 + lr;                        // N index of this lane
  float bias = bb1[col];
  float* c0  = H1 + (tm * 16 + 8 * half) * 128 + col;
#pragma unroll
  for (int v = 0; v < 8; ++v) {                     // M = v + 8*half
    float val = acc[v] + bias;
    c0[v * 128] = val > 0.f ? val : 0.f;
  }
}

// ---- K3: rank-1 attention + w_out + relu ----------------------------------
// qkv[i,n,r] = h[i,n]*rowsum(w_in[r]) + b_in[r]  (broadcast => affine in t).
// softmax(q_i k_j) over j == softmax(s_i t_j) with s_i = q_i*alpha_k.
// One block per feature column n; t[512] staged in LDS (broadcast reads).
__global__ void attn_kernel(const float* __restrict__ H1,
                            const float* __restrict__ w_in,
                            const float* __restrict__ b_in,
                            const float* __restrict__ w_out,
                            const float* __restrict__ b_out,
                            float* __restrict__ F) {
  __shared__ float sT[512];
  __shared__ float sMin[256];
  __shared__ float sMax[256];
  int n   = blockIdx.x;       // 0..127
  int tid = threadIdx.x;      // 0..255
  float t0 = H1[tid * 128 + n];
  float t1 = H1[(tid + 256) * 128 + n];
  sT[tid] = t0; sT[tid + 256] = t1;
  sMin[tid] = fminf(t0, t1);
  sMax[tid] = fmaxf(t0, t1);
  __syncthreads();
  for (int st = 128; st > 0; st >>= 1) {
    if (tid < st) {
      sMin[tid] = fminf(sMin[tid], sMin[tid + st]);
      sMax[tid] = fmaxf(sMax[tid], sMax[tid + st]);
    }
    __syncthreads();
  }
  float tmin = sMin[0], tmax = sMax[0];

  float sw[12];
#pragma unroll
  for (int r = 0; r < 12; ++r)
    sw[r] = w_in[r*4+0] + w_in[r*4+1] + w_in[r*4+2] + w_in[r*4+3];

  for (int pass = 0; pass < 2; ++pass) {
    int i = tid + pass * 256;
    float ti = sT[i];
    float s[4], shift[4], den[4], num[4];
#pragma unroll
    for (int h = 0; h < 4; ++h) {
      float q  = fmaf(sw[h], ti, b_in[h]);   // q_i
      s[h]     = q * sw[4 + h];              // q_i * alpha_k (beta_k cancels)
      shift[h] = s[h] * (s[h] >= 0.f ? tmax : tmin);  // stabilizer
      den[h] = 0.f; num[h] = 0.f;
    }
    for (int j = 0; j < 512; ++j) {
      float tj = sT[j];                      // LDS broadcast
#pragma unroll
      for (int h = 0; h < 4; ++h) {
        float e = __expf(fmaf(s[h], tj, -shift[h]));
        den[h] += e;
        num[h] = fmaf(e, tj, num[h]);
      }
    }
    float o[4];
#pragma unroll
    for (int h = 0; h < 4; ++h)             // o = alpha_v * E[t] + beta_v
      o[h] = fmaf(sw[8 + h], num[h] / den[h], b_in[8 + h]);
#pragma unroll
    for (int r = 0; r < 4; ++r) {           // o @ w_out^T + b_out, relu
      float val = b_out[r];
#pragma unroll
      for (int h = 0; h < 4; ++h) val = fmaf(w_out[r * 4 + h], o[h], val);
      F[i * 512 + n * 4 + r] = val > 0.f ? val : 0.f;
    }
  }
}

// ---- K4: batchnorm1 over batch axis ---------------------------------------
__global__ void bn1_normalize(const float* __restrict__ F,
                              const float* __restrict__ g,
                              const float* __restrict__ b,
                              float* __restrict__ FN) {
  int c = blockIdx.x * blockDim.x + threadIdx.x;  // 0..511
  float s = 0.f, s2 = 0.f;
  for (int r = 0; r < 512; ++r) {
    float v = F[r * 512 + c];
    s += v; s2 += v * v;
  }
  float m   = s  * (1.f / 512.f);
  float var = s2 * (1.f / 512.f) - m * m;
  float sc  = g[c] * rsqrtf(var + EPS);
  float bi  = b[c] - m * sc;
  for (int r = 0; r < 512; ++r)
    FN[r * 512 + c] = F[r * 512 + c] * sc + bi;
}

// ---- K5: GEMM2 via V_WMMA_F32_16X16X4_F32, fused bias+relu ----------------
// H2(512x32) = relu( FN(512x512) @ w2^T(512x32) + bb2 ); 32x2 = 64 tiles.
__global__ void gemm2_wmma(const float* __restrict__ FN,
                           const float* __restrict__ w2,
                           const float* __restrict__ bb2,
                           float* __restrict__ H2) {
  int wave = (blockIdx.x * blockDim.x + threadIdx.x) >> 5;
  int lane = threadIdx.x & 31;
  int half = lane >> 4;
  int lr   = lane & 15;
  int tm   = wave >> 1;          // 0..31
  int tn   = wave & 1;           // 0..1
  const float* arow = FN + (tm * 16 + lr) * 512;
  const float* brow = w2 + (tn * 16 + lr) * 512;  // row of w2 = col of w2^T
  v8f acc = {};
  for (int kk = 0; kk < 512; kk += 4) {
    int k2 = kk + 2 * half;
    v2f a  = *reinterpret_cast<const v2f*>(arow + k2);
    v2f bv = *reinterpret_cast<const v2f*>(brow + k2);
    acc = __builtin_amdgcn_wmma_f32_16x16x4_f32(false, a, false, bv,
                                                (short)0, acc, false, false);
  }
  int col    = tn * 16 + lr;
  float bias = bb2[col];
  float* c0  = H2 + (tm * 16 + 8 * half) * 32 + col;
#pragma unroll
  for (int v = 0; v < 8; ++v) {
    float val = acc[v] + bias;
    c0[v * 32] = val > 0.f ? val : 0.f;
  }
}

// ---- K6: tiny tail GEMM, (512x32)@(32x2)+bb3 ------------------------------
__global__ void gemm3(const float* __restrict__ H2,
                      const float* __restrict__ w3,
                      const float* __restrict__ bb3,
                      float* __restrict__ out) {
  int r = blockIdx.x * blockDim.x + threadIdx.x;
  if (r >= 512) return;
  float a0 = bb3[0], a1 = bb3[1];
#pragma unroll
  for (int k = 0; k < 32; ++k) {
    float h = H2[r * 32 + k];
    a0 = fmaf(h, w3[k],      a0);
    a1 = fmaf(h, w3[32 + k], a1);
  }
  out[r * 2 + 0] = a0;
  out[r * 2 + 1] = a1;
}

extern "C" void kernel_launch(void* const* d_in, const int* in_sizes, int n_in,
                              void* d_out, int out_size, void* d_ws, size_t ws_size,
                              hipStream_t stream) {
  (void)in_sizes; (void)n_in; (void)out_size; (void)ws_size;
  const float* x     = (const float*)d_in[0];
  const float* bn0g  = (const float*)d_in[1];
  const float* bn0b  = (const float*)d_in[2];
  const float* w1    = (const float*)d_in[3];
  const float* bb1   = (const float*)d_in[4];
  const float* w_in  = (const float*)d_in[5];
  const float* b_in  = (const float*)d_in[6];
  const float* w_out = (const float*)d_in[7];
  const float* b_out = (const float*)d_in[8];
  const float* bn1g  = (const float*)d_in[9];
  const float* bn1b  = (const float*)d_in[10];
  const float* w2    = (const float*)d_in[11];
  const float* bb2   = (const float*)d_in[12];
  const float* w3    = (const float*)d_in[13];
  const float* bb3   = (const float*)d_in[14];

  float* ws = (float*)d_ws;
  float* XN = ws + XN_OFF;
  float* H1 = ws + H1_OFF;
  float* F  = ws + F_OFF;
  float* FN = ws + FN_OFF;
  float* H2 = ws + H2_OFF;
  float* out = (float*)d_out;

  bn0_normalize<<<3,   256, 0, stream>>>(x, bn0g, bn0b, XN);
  gemm1_wmma  <<<32,  256, 0, stream>>>(XN, w1, bb1, H1);
  attn_kernel <<<128, 256, 0, stream>>>(H1, w_in, b_in, w_out, b_out, F);
  bn1_normalize<<<2,  256, 0, stream>>>(F, bn1g, bn1b, FN);
  gemm2_wmma  <<<8,   256, 0, stream>>>(FN, w2, bb2, H2);
  gemm3       <<<4,   128, 0, stream>>>(H2, w3, bb3, out);
}